// RecurrentGCN_45758581572160
// MI455X (gfx1250) — compile-verified
//
#include <hip/hip_runtime.h>
#include <hip/hip_bf16.h>

typedef __attribute__((ext_vector_type(2))) float v2f;
typedef __attribute__((ext_vector_type(8))) float v8f;

#define HID 64
#define INDIM 128
#define KDIM 320            // 128 (x) + 3*64 (T0,T1,T2)
#define GCOLS 256           // 4 gates * 64
#define LMAX 2.0f

__device__ __forceinline__ v8f wmma_f32_4(v2f a, v2f b, v8f c) {
  // D = A(16x4 f32) * B(4x16 f32) + C(16x16 f32)
  return __builtin_amdgcn_wmma_f32_16x16x4_f32(false, a, false, b, (short)0, c, false, false);
}

__device__ __forceinline__ float fsigmoid(float x) {
  return 1.0f / (1.0f + __expf(-x));
}

// Native no-return fp32 scatter-add: executes in the L2 atomic units, tracked
// on STOREcnt (no VGPR writeback), device scope for cross-WGP visibility.
__device__ __forceinline__ void atomic_fadd(float* p, float v) {
  asm volatile("global_atomic_add_f32 %0, %1, off scope:SCOPE_DEV"
               :: "v"(p), "v"(v) : "memory");
}

// ---------------- sparse preprocessing ----------------

__global__ __launch_bounds__(256)
void k_degree(const int* __restrict__ src, const int* __restrict__ dst,
              const float* __restrict__ ew, float* __restrict__ deg, int E) {
  int e = blockIdx.x * 256 + threadIdx.x;
  if (e >= E) return;
  int s = src[e], d = dst[e];
  if (s != d) atomic_fadd(&deg[s], ew[e]);
}

__global__ __launch_bounds__(256)
void k_dinv(const float* __restrict__ deg, float* __restrict__ dinv, int Nn) {
  int i = blockIdx.x * 256 + threadIdx.x;
  if (i >= Nn) return;
  float dg = deg[i];
  dinv[i] = (dg > 0.0f) ? rsqrtf(dg) : 0.0f;
}

__global__ __launch_bounds__(256)
void k_lapw(const int* __restrict__ src, const int* __restrict__ dst,
            const float* __restrict__ ew, const float* __restrict__ dinv,
            float* __restrict__ lapw, int E) {
  int e = blockIdx.x * 256 + threadIdx.x;
  if (e >= E) return;
  int s = src[e], d = dst[e];
  float w = (s == d) ? 0.0f : ew[e];
  lapw[e] = (2.0f / LMAX) * (-dinv[s] * w * dinv[d]);
}

// dst[i] = scale * src[i]   (diag term of L_hat; scale = 2/lmax - 1)
__global__ __launch_bounds__(256)
void k_scale(float* __restrict__ dstp, const float* __restrict__ srcp,
             float scale, int count) {
  int i = blockIdx.x * 256 + threadIdx.x;
  if (i >= count) return;
  dstp[i] = scale * srcp[i];
}

// acc[dst] += lap_w[e] * z[src] : 16 threads per edge, float4 gather per thread,
// 4 native f32 atomics into the L2-resident accumulator.
__global__ __launch_bounds__(256)
void k_spmm(const float* __restrict__ z, float* __restrict__ acc,
            const int* __restrict__ src, const int* __restrict__ dst,
            const float* __restrict__ lw, int E) {
  int gid = blockIdx.x * 256 + threadIdx.x;
  int e = gid >> 4;
  if (e >= E) return;
  float w = lw[e];
  if (w == 0.0f) return;
  int ch = (gid & 15) << 2;
  int s = src[e], d = dst[e];
  const float4 zv = *(const float4*)(z + (size_t)s * HID + ch);
  float* out = acc + (size_t)d * HID + ch;
  atomic_fadd(out + 0, w * zv.x);
  atomic_fadd(out + 1, w * zv.y);
  atomic_fadd(out + 2, w * zv.z);
  atomic_fadd(out + 3, w * zv.w);
}

// T2 = 2*T2 - h   (Chebyshev: T2 = 2*Lhat*T1 - T0)
__global__ __launch_bounds__(256)
void k_cheb2(float* __restrict__ t2, const float* __restrict__ h, int count) {
  int i = blockIdx.x * 256 + threadIdx.x;
  if (i >= count) return;
  t2[i] = 2.0f * t2[i] - h[i];
}

// ---------------- pack B = [Wx | theta_k] into [320][256], bias[256] ----------------

__global__ __launch_bounds__(256)
void k_packB(const float* __restrict__ Wx, const float* __restrict__ theta,
             const float* __restrict__ bg, const float* __restrict__ conv_b,
             float* __restrict__ Bp, float* __restrict__ bias) {
  int gid = blockIdx.x * 256 + threadIdx.x;
  if (gid >= KDIM * GCOLS) return;
  int k = gid / GCOLS, col = gid % GCOLS;
  int g = col >> 6, o = col & 63;
  float v;
  if (k < INDIM) {
    v = Wx[((size_t)g * INDIM + k) * HID + o];
  } else {
    int kk = k - INDIM;
    int kb = kk >> 6, kr = kk & 63;
    v = theta[(((size_t)g * 3 + kb) * HID + kr) * HID + o];
  }
  Bp[gid] = v;
  if (gid < GCOLS) bias[gid] = bg[g * HID + o] + conv_b[g * HID + o];
}

// ---------------- fused gate GEMM [N,320]x[320,256] + LSTM epilogue ----------------

#define AS1 324   // LDS stride for A tile (16 x 320)
#define PS1 260   // LDS stride for pre tile (16 x 256)

__global__ __launch_bounds__(256)
void k_gemm_gates(const float* __restrict__ x, const float* __restrict__ h,
                  const float* __restrict__ T1, const float* __restrict__ T2,
                  const float* __restrict__ Bp, const float* __restrict__ bias,
                  const float* __restrict__ c_in,
                  float* __restrict__ h_out, float* __restrict__ c_out, int Nn) {
  __shared__ float Asmem[16 * AS1];
  __shared__ float Psmem[16 * PS1];
  const int tid = threadIdx.x;
  const int m0 = blockIdx.x * 16;

  // cooperative A tile load: 16 rows, cols [0,128)=x, [128,192)=h, [192,256)=T1, [256,320)=T2
  for (int i = tid; i < 16 * KDIM; i += 256) {
    int r = i / KDIM, k = i % KDIM;
    int n = m0 + r; if (n >= Nn) n = Nn - 1;
    float v;
    if (k < 128)      v = x[(size_t)n * INDIM + k];
    else if (k < 192) v = h[(size_t)n * HID + (k - 128)];
    else if (k < 256) v = T1[(size_t)n * HID + (k - 192)];
    else              v = T2[(size_t)n * HID + (k - 256)];
    Asmem[r * AS1 + k] = v;
  }
  __syncthreads();

  const int wave = tid >> 5, lane = tid & 31;
  const int lo = lane & 15, hi = lane >> 4;
  const int c0 = wave * 32;   // 8 waves * 32 cols = 256 cols
  v8f acc0 = {}; v8f acc1 = {};

  for (int k0 = 0; k0 < KDIM; k0 += 4) {
    // A frag: lane holds M=lo, K = k0 + 2*hi + {0,1}
    v2f a = *(const v2f*)&Asmem[lo * AS1 + k0 + 2 * hi];
    // B frag: lane holds N=lo, K = k0 + 2*hi + {0,1}
    const float* bptr = Bp + (size_t)(k0 + 2 * hi) * GCOLS + c0 + lo;
    v2f b0, b1;
    b0[0] = bptr[0];    b0[1] = bptr[GCOLS];
    b1[0] = bptr[16];   b1[1] = bptr[GCOLS + 16];
    acc0 = wmma_f32_4(a, b0, acc0);
    acc1 = wmma_f32_4(a, b1, acc1);
  }

  float bias0 = bias[c0 + lo], bias1 = bias[c0 + 16 + lo];
#pragma unroll
  for (int r = 0; r < 8; ++r) {
    int row = r + 8 * hi;            // C/D layout: vgpr r -> M = r + 8*(lane/16)
    Psmem[row * PS1 + c0 + lo]      = acc0[r] + bias0;
    Psmem[row * PS1 + c0 + 16 + lo] = acc1[r] + bias1;
  }
  __syncthreads();

  // LSTM gates: pre order [i, f, t, o] in column blocks of 64
#pragma unroll
  for (int j = 0; j < 4; ++j) {
    int flat = j * 256 + tid;
    int row = flat >> 6, o = flat & 63;
    int n = m0 + row;
    if (n >= Nn) continue;
    const float* pr = &Psmem[row * PS1];
    float ig = fsigmoid(pr[o]);
    float fg = fsigmoid(pr[64 + o]);
    float tg = tanhf(pr[128 + o]);
    float og = fsigmoid(pr[192 + o]);
    float cv = fg * c_in[(size_t)n * HID + o] + ig * tg;
    float hv = og * tanhf(cv);
    h_out[(size_t)n * HID + o] = hv;
    c_out[(size_t)n * HID + o] = cv;
  }
}

// ---------------- output GEMM: out = relu(h0) @ Wl + bl ----------------

#define AS2 68
#define BS2 68

__global__ __launch_bounds__(256)
void k_gemm_out(const float* __restrict__ h0, const float* __restrict__ Wl,
                const float* __restrict__ bl, float* __restrict__ out, int Nn) {
  __shared__ float Bs[64 * BS2];
  __shared__ float As[32 * AS2];
  const int tid = threadIdx.x;
  const int m0 = blockIdx.x * 32;

  for (int i = tid; i < 64 * 64; i += 256) {
    int k = i >> 6, ccol = i & 63;
    Bs[k * BS2 + ccol] = Wl[i];
  }
  for (int i = tid; i < 32 * 64; i += 256) {
    int r = i >> 6, k = i & 63;
    int n = m0 + r; if (n >= Nn) n = Nn - 1;
    float v = h0[(size_t)n * HID + k];
    As[r * AS2 + k] = v > 0.0f ? v : 0.0f;     // trailing ReLU
  }
  __syncthreads();

  const int wave = tid >> 5, lane = tid & 31;
  const int lo = lane & 15, hi = lane >> 4;
  const int r0 = (wave >> 2) * 16, c0 = (wave & 3) * 16;   // 2x4 waves -> 32x64 tile
  v8f acc = {};
  for (int k0 = 0; k0 < 64; k0 += 4) {
    v2f a = *(const v2f*)&As[(r0 + lo) * AS2 + k0 + 2 * hi];
    v2f b;
    b[0] = Bs[(k0 + 2 * hi) * BS2 + c0 + lo];
    b[1] = Bs[(k0 + 2 * hi + 1) * BS2 + c0 + lo];
    acc = wmma_f32_4(a, b, acc);
  }
  float bb = bl[c0 + lo];
#pragma unroll
  for (int r = 0; r < 8; ++r) {
    int row = r0 + r + 8 * hi;
    int n = m0 + row;
    if (n < Nn) out[(size_t)n * HID + c0 + lo] = acc[r] + bb;
  }
}

// ---------------- launcher ----------------

extern "C" void kernel_launch(void* const* d_in, const int* in_sizes, int n_in,
                              void* d_out, int out_size, void* d_ws, size_t ws_size,
                              hipStream_t stream) {
  const float* x      = (const float*)d_in[0];
  const int*   ei     = (const int*)d_in[1];
  const float* ew     = (const float*)d_in[2];
  const float* h      = (const float*)d_in[3];
  const float* c      = (const float*)d_in[4];
  const float* Wx     = (const float*)d_in[5];
  const float* bg     = (const float*)d_in[6];
  const float* theta  = (const float*)d_in[7];
  const float* conv_b = (const float*)d_in[8];
  const float* Wl     = (const float*)d_in[9];
  const float* bl     = (const float*)d_in[10];

  const int E  = in_sizes[2];
  const int Nn = in_sizes[3] / HID;
  const int* src = ei;
  const int* dst = ei + E;

  // workspace layout (floats)
  float* deg  = (float*)d_ws;                          // N
  float* dinv = deg  + Nn;                             // N
  float* lapw = dinv + Nn;                             // E
  float* T1   = lapw + E;                              // N*64
  float* T2   = T1 + (size_t)Nn * HID;                 // N*64
  float* Bp   = T2 + (size_t)Nn * HID;                 // 320*256
  float* bias = Bp + (size_t)KDIM * GCOLS;             // 256

  float* out   = (float*)d_out;                        // N*64
  float* h_out = out   + (size_t)Nn * HID;             // N*64
  float* c_out = h_out + (size_t)Nn * HID;             // N*64

  const float DIAG = 2.0f / LMAX - 1.0f;               // 0 at lambda_max = 2
  const int NH = Nn * HID;

  hipMemsetAsync(deg, 0, (size_t)Nn * sizeof(float), stream);
  k_degree<<<(E + 255) / 256, 256, 0, stream>>>(src, dst, ew, deg, E);
  k_dinv<<<(Nn + 255) / 256, 256, 0, stream>>>(deg, dinv, Nn);
  k_lapw<<<(E + 255) / 256, 256, 0, stream>>>(src, dst, ew, dinv, lapw, E);

  // T1 = Lhat @ h
  k_scale<<<(NH + 255) / 256, 256, 0, stream>>>(T1, h, DIAG, NH);
  k_spmm<<<(E * 16 + 255) / 256, 256, 0, stream>>>(h, T1, src, dst, lapw, E);
  // T2 = 2 * (Lhat @ T1) - h
  k_scale<<<(NH + 255) / 256, 256, 0, stream>>>(T2, T1, DIAG, NH);
  k_spmm<<<(E * 16 + 255) / 256, 256, 0, stream>>>(T1, T2, src, dst, lapw, E);
  k_cheb2<<<(NH + 255) / 256, 256, 0, stream>>>(T2, h, NH);

  k_packB<<<(KDIM * GCOLS + 255) / 256, 256, 0, stream>>>(Wx, theta, bg, conv_b, Bp, bias);

  k_gemm_gates<<<(Nn + 15) / 16, 256, 0, stream>>>(x, h, T1, T2, Bp, bias, c,
                                                   h_out, c_out, Nn);
  k_gemm_out<<<(Nn + 31) / 32, 256, 0, stream>>>(h_out, Wl, bl, out, Nn);
}